// SoftLabelCrossEntropyLoss_14800457302459
// MI455X (gfx1250) — compile-verified
//
#include <hip/hip_runtime.h>

typedef __attribute__((ext_vector_type(2))) float v2f;
typedef __attribute__((ext_vector_type(8))) float v8f;

#define NCLS 11
#define LW 0.15f
#define RW 0.15f
#define CW 0.7f

// Select x[i] from 11 register-resident values (unrolled cndmask chain; no scratch).
__device__ __forceinline__ float gather11(const float* x, int i) {
  float r = x[0];
#pragma unroll
  for (int k = 1; k < NCLS; ++k) r = (i == k) ? x[k] : r;
  return r;
}

// loss_row = logZ - sum(normalized_w * x_at_idx); weights sum to 1 after normalization.
__device__ __forceinline__ float row_loss(const float* x, int t) {
  float m = x[0];
#pragma unroll
  for (int k = 1; k < NCLS; ++k) m = fmaxf(m, x[k]);
  float s = 0.f;
#pragma unroll
  for (int k = 0; k < NCLS; ++k) s += __expf(x[k] - m);
  float logZ = m + __logf(s);

  int tl = (t > 0) ? t - 1 : 0;
  int tr = (t < NCLS - 1) ? t + 1 : NCLS - 1;
  float xl = gather11(x, tl);
  float xt = gather11(x, t);
  float xr = gather11(x, tr);

  float wl = (t == 0) ? 0.f : LW;
  float wr = (t == NCLS - 1) ? 0.f : RW;
  float wn = (t == 0 || t == NCLS - 1) ? (1.0f / (CW + LW)) : 1.0f;
  return logZ - (wl * xl + CW * xt + wr * xr) * wn;
}

// Stage 1: each thread streams 4-row (176B, 16B-aligned) chunks via 11x b128 loads,
// accumulates per-thread loss, wave-reduces with V_WMMA_F32_16X16X4_F32, writes
// one partial per block.
__global__ void __launch_bounds__(256) slce_main(const float* __restrict__ logits,
                                                 const int* __restrict__ target,
                                                 float* __restrict__ partials,
                                                 int nchunks) {
  const int tid = blockIdx.x * blockDim.x + threadIdx.x;
  const int stride = gridDim.x * blockDim.x;

  float acc = 0.f;
  for (int c = tid; c < nchunks; c += stride) {
    const float4* p = (const float4*)logits + (size_t)c * 11;
    if (c + stride < nchunks) {
      // gfx1250 global_prefetch_b8 on the next grid-stride chunk
      __builtin_prefetch((const void*)((const float4*)logits + (size_t)(c + stride) * 11), 0, 1);
    }
    float f[44];
#pragma unroll
    for (int j = 0; j < 11; ++j) {
      float4 v = p[j];
      f[4 * j + 0] = v.x;
      f[4 * j + 1] = v.y;
      f[4 * j + 2] = v.z;
      f[4 * j + 3] = v.w;
    }
    int4 tv = *((const int4*)target + c);
    acc += row_loss(&f[0], tv.x);
    acc += row_loss(&f[11], tv.y);
    acc += row_loss(&f[22], tv.z);
    acc += row_loss(&f[33], tv.w);
  }

  // Wave reduction via WMMA: A(16x4) holds the 32 partials (lanes 0-15 -> K0,
  // lanes 16-31 -> K2, other K zero), B = all ones (layout-free since every
  // element is 1.0).  D[m][n] = p[m] + p[m+16].  EXEC is all-1s here.
  v2f a;
  a.x = acc;
  a.y = 0.f;
  v2f b;
  b.x = 1.f;
  b.y = 1.f;
  v8f cz = {};
  v8f d = __builtin_amdgcn_wmma_f32_16x16x4_f32(false, a, false, b, (short)0, cz, false, false);
  float s = d[0] + d[1] + d[2] + d[3] + d[4] + d[5] + d[6] + d[7];
  // lane L in [0,16) holds sum over lanes {0..7,16..23}; lane L+16 holds the rest.

  __shared__ float sh[16];  // 8 waves x 2 column-halves
  const int lane = threadIdx.x & 31;
  const int wv = threadIdx.x >> 5;
  if (lane == 0) sh[2 * wv] = s;
  if (lane == 16) sh[2 * wv + 1] = s;
  __syncthreads();
  if (threadIdx.x == 0) {
    float bs = 0.f;
#pragma unroll
    for (int i = 0; i < 16; ++i) bs += sh[i];
    partials[blockIdx.x] = bs;
  }
}

// Stage 2: deterministic single-block reduction of block partials (double accum),
// plus serial handling of any tail rows (B % 4), then mean.
__global__ void __launch_bounds__(256) slce_reduce(const float* __restrict__ partials, int nparts,
                                                   const float* __restrict__ logits,
                                                   const int* __restrict__ target,
                                                   int tail_start, int B,
                                                   float* __restrict__ out) {
  __shared__ double sh[256];
  double s = 0.0;
  for (int i = threadIdx.x; i < nparts; i += blockDim.x) s += (double)partials[i];
  if (threadIdx.x == 0) {
    for (int r = tail_start; r < B; ++r) {
      float x[NCLS];
#pragma unroll
      for (int k = 0; k < NCLS; ++k) x[k] = logits[(size_t)r * NCLS + k];
      s += (double)row_loss(x, target[r]);
    }
  }
  sh[threadIdx.x] = s;
  __syncthreads();
  for (int off = 128; off > 0; off >>= 1) {
    if (threadIdx.x < (unsigned)off) sh[threadIdx.x] += sh[threadIdx.x + off];
    __syncthreads();
  }
  if (threadIdx.x == 0) out[0] = (float)(sh[0] / (double)B);
}

extern "C" void kernel_launch(void* const* d_in, const int* in_sizes, int n_in,
                              void* d_out, int out_size, void* d_ws, size_t ws_size,
                              hipStream_t stream) {
  const float* logits = (const float*)d_in[0];
  const int* target = (const int*)d_in[1];
  const int B = in_sizes[1];

  const int nchunks = B >> 2;        // 4 rows per chunk
  const int tail_start = nchunks << 2;

  float* partials = (float*)d_ws;

  const int threads = 256;
  int blocks = (nchunks + threads - 1) / threads;
  if (blocks > 2048) blocks = 2048;
  if (blocks < 1) blocks = 1;

  slce_main<<<blocks, threads, 0, stream>>>(logits, target, partials, nchunks);
  slce_reduce<<<1, 256, 0, stream>>>(partials, blocks, logits, target, tail_start, B,
                                     (float*)d_out);
}